// GraphEncoder_24653112279422
// MI455X (gfx1250) — compile-verified
//
#include <hip/hip_runtime.h>
#include <hip/hip_bf16.h>

typedef float v2f __attribute__((ext_vector_type(2)));
typedef float v8f __attribute__((ext_vector_type(8)));

#define N_NODES 100000

// ---- init: deg = 1.0 (self loop counted analytically) ----
__global__ void k_init_deg(float* deg, int n) {
  int i = blockIdx.x * blockDim.x + threadIdx.x;
  if (i < n) deg[i] = 1.0f;
}

// ---- init accumulator rows with bias (fm = featdim-1, power-of-two mask) ----
__global__ void k_init_bias(float* out, const float* __restrict__ b, int total, int fm) {
  int i = blockIdx.x * blockDim.x + threadIdx.x;
  if (i < total) out[i] = b[i & fm];
}

// ---- degree histogram over dst ----
__global__ void k_degree(const int* __restrict__ dst, float* deg, int E) {
  int e = blockIdx.x * blockDim.x + threadIdx.x;
  if (e < E) atomicAdd(&deg[dst[e]], 1.0f);
}

// ---- inv = rsqrt(deg), in place (deg >= 1 always due to self loops) ----
__global__ void k_invsqrt(float* deg, int n) {
  int i = blockIdx.x * blockDim.x + threadIdx.x;
  if (i < n) deg[i] = rsqrtf(deg[i]);
}

// ---- fp32 WMMA GEMM, compile-time dims: C[M,N] = A[M,K] @ B[K,N] ----
// One 16x16 tile per wave. 16x16x4 f32 fragment layout:
//   A: 2 VGPRs (lanes 0-15: K=k,k+1 ; lanes 16-31: K=k+2,k+3), rows across lanes
//   C: 8 VGPRs (VGPR i: lanes 0-15 -> M=i, lanes 16-31 -> M=8+i)
// Constexpr N/K -> every load/store offset is an immediate; full unroll -> 32
// back-to-back v_wmma with clause-batched loads, no per-iteration address math.
template <int N, int K>
__global__ __launch_bounds__(256) void k_gemm_wmma(
    const float* __restrict__ A, const float* __restrict__ B, float* __restrict__ C,
    int totalTiles) {
  constexpr int tilesN = N / 16;
  int wave = threadIdx.x >> 5;
  int lane = threadIdx.x & 31;
  int tid  = blockIdx.x * (blockDim.x >> 5) + wave;
  if (tid >= totalTiles) return;           // wave-uniform: EXEC stays all-1s
  int tm = tid / tilesN;                   // tilesN is power of two -> shift
  int tn = tid - tm * tilesN;
  int half = lane >> 4;                    // 0: K offset 0/1, 1: K offset 2/3
  int l16  = lane & 15;

  const float* arow = A + (size_t)(tm * 16 + l16) * K + half * 2;
  const float* bcol = B + (size_t)(half * 2) * N + tn * 16 + l16;

  v8f c = {0.f, 0.f, 0.f, 0.f, 0.f, 0.f, 0.f, 0.f};
#pragma unroll
  for (int k = 0; k < K; k += 4) {
    v2f a = *(const v2f*)(arow + k);       // global_load_b64, imm offset k*4
    v2f b;
    b.x = bcol[k * N];                     // imm offsets, same base
    b.y = bcol[k * N + N];
    c = __builtin_amdgcn_wmma_f32_16x16x4_f32(
        /*neg_a=*/false, a, /*neg_b=*/false, b,
        /*c_mod=*/(short)0, c, /*reuse_a=*/false, /*reuse_b=*/false);
  }

  float* crow = C + (size_t)(tm * 16 + half * 8) * N + tn * 16 + l16;
#pragma unroll
  for (int i = 0; i < 8; ++i) crow[i * N] = c[i];
}

// ---- edge scatter, 128 feats: one wave per edge, float4 per lane ----
__global__ __launch_bounds__(256) void k_agg128(
    const float* __restrict__ h, const float* __restrict__ inv,
    const int* __restrict__ src, const int* __restrict__ dst,
    float* out, int E) {
  int e = blockIdx.x * (blockDim.x >> 5) + (threadIdx.x >> 5);
  if (e >= E) return;
  int lane = threadIdx.x & 31;
  int s = src[e], d = dst[e];
  float nrm = inv[s] * inv[d];
  float4 v = ((const float4*)(h + (size_t)s * 128))[lane];
  float* o = out + (size_t)d * 128 + lane * 4;
  atomicAdd(o + 0, v.x * nrm);
  atomicAdd(o + 1, v.y * nrm);
  atomicAdd(o + 2, v.z * nrm);
  atomicAdd(o + 3, v.w * nrm);
}

// ---- edge scatter, 64 feats: one wave per edge, float2 per lane ----
__global__ __launch_bounds__(256) void k_agg64(
    const float* __restrict__ h, const float* __restrict__ inv,
    const int* __restrict__ src, const int* __restrict__ dst,
    float* out, int E) {
  int e = blockIdx.x * (blockDim.x >> 5) + (threadIdx.x >> 5);
  if (e >= E) return;
  int lane = threadIdx.x & 31;
  int s = src[e], d = dst[e];
  float nrm = inv[s] * inv[d];
  float2 v = ((const float2*)(h + (size_t)s * 64))[lane];
  float* o = out + (size_t)d * 64 + lane * 2;
  atomicAdd(o + 0, v.x * nrm);
  atomicAdd(o + 1, v.y * nrm);
}

// ---- self-loop add (norm = inv[i]^2) fused with ReLU (layer1 epilogue) ----
__global__ void k_self_relu(float* agg, const float* __restrict__ h,
                            const float* __restrict__ inv, int total) {
  int i = blockIdx.x * blockDim.x + threadIdx.x;
  if (i >= total) return;
  int node = i >> 7;                       // 128 feats
  float w = inv[node];
  float v = agg[i] + h[i] * w * w;
  agg[i] = fmaxf(v, 0.0f);
}

// ---- self-loop add, layer2 epilogue (bias already seeded) ----
__global__ void k_self2(float* out, const float* __restrict__ h,
                        const float* __restrict__ inv, int total) {
  int i = blockIdx.x * blockDim.x + threadIdx.x;
  if (i >= total) return;
  int node = i >> 6;                       // 64 feats
  float w = inv[node];
  out[i] += h[i] * w * w;
}

extern "C" void kernel_launch(void* const* d_in, const int* in_sizes, int n_in,
                              void* d_out, int out_size, void* d_ws, size_t ws_size,
                              hipStream_t stream) {
  (void)n_in; (void)out_size; (void)ws_size;
  const float* x  = (const float*)d_in[0];
  const int*   ei = (const int*)d_in[1];
  const float* W1 = (const float*)d_in[2];
  const float* b1 = (const float*)d_in[3];
  const float* W2 = (const float*)d_in[4];
  const float* b2 = (const float*)d_in[5];
  float* dout = (float*)d_out;

  const int n = N_NODES;
  const int E = in_sizes[1] / 2;
  const int* src = ei;        // edge_index[0]
  const int* dst = ei + E;    // edge_index[1]

  // workspace carve-out (floats): inv[n] | h1[n*128] | agg1[n*128] | h2[n*64]
  float* ws   = (float*)d_ws;
  float* inv  = ws;
  float* h1   = inv + n;
  float* agg1 = h1 + (size_t)n * 128;
  float* h2   = agg1 + (size_t)n * 128;

  const int T = 256;
  // init
  k_init_deg <<<(n + T - 1) / T, T, 0, stream>>>(inv, n);
  k_init_bias<<<(n * 128 + T - 1) / T, T, 0, stream>>>(agg1, b1, n * 128, 127);
  k_init_bias<<<(n * 64  + T - 1) / T, T, 0, stream>>>(dout, b2, n * 64, 63);
  // normalization
  k_degree  <<<(E + T - 1) / T, T, 0, stream>>>(dst, inv, E);
  k_invsqrt <<<(n + T - 1) / T, T, 0, stream>>>(inv, n);
  // layer 1: GEMM (100000x128 @ 128x128) + aggregate + bias/relu epilogue
  {
    int total = (n / 16) * (128 / 16);                 // 50000 tiles
    k_gemm_wmma<128, 128><<<(total + 7) / 8, T, 0, stream>>>(x, W1, h1, total);
  }
  k_agg128   <<<(E + 7) / 8, T, 0, stream>>>(h1, inv, src, dst, agg1, E);
  k_self_relu<<<(n * 128 + T - 1) / T, T, 0, stream>>>(agg1, h1, inv, n * 128);
  // layer 2: GEMM (100000x128 @ 128x64) + aggregate + self-loop epilogue
  {
    int total = (n / 16) * (64 / 16);                  // 25000 tiles
    k_gemm_wmma<64, 128><<<(total + 7) / 8, T, 0, stream>>>(agg1, W2, h2, total);
  }
  k_agg64 <<<(E + 7) / 8, T, 0, stream>>>(h2, inv, src, dst, dout, E);
  k_self2 <<<(n * 64 + T - 1) / T, T, 0, stream>>>(dout, h2, inv, n * 64);
}